// OffsetsSDF_89790586290709
// MI455X (gfx1250) — compile-verified
//
#include <hip/hip_runtime.h>

typedef __attribute__((ext_vector_type(2))) float v2f;
typedef __attribute__((ext_vector_type(8))) float v8f;

#define L_LEVELS   24
#define T_SIZE     (1u << 19)
#define N_PTS      524288
#define ENC_DIM    48
#define ENC_STRIDE 50   // even -> 8B-aligned float pairs, spreads LDS banks
#define H_STRIDE   34
#define WAVES      4
#define BLOCK      (WAVES * 32)
#define MIN_OFF    1.0e-4f

__device__ __forceinline__ float softplusf(float x) {
    // numerically stable: max(x,0) + log1p(exp(-|x|))
    return fmaxf(x, 0.0f) + log1pf(expf(-fabsf(x)));
}

// One 16x16 f32 output tile of  A(16xK, LDS) * B(Kx16, global), K multiple of 4.
// A layout per ISA 7.12.2 (fp32 16x4): lane l holds row M=l&15; lanes0-15 K=k0,k0+1;
// lanes16-31 K=k0+2,k0+3.  B mirrored: row K striped the same way, col = l&15.
__device__ __forceinline__ v8f gemm_tile_f32(const float* __restrict__ aLDS, int aStride,
                                             int K, const float* __restrict__ B, int ldB,
                                             int colBase, int maxCol, int lane) {
    const int nl = lane & 15;
    const int kh = lane >> 4;
    const int col = colBase + nl;
    const bool inb = (col < maxCol);
    v8f acc = {0.f, 0.f, 0.f, 0.f, 0.f, 0.f, 0.f, 0.f};
#pragma unroll
    for (int k0 = 0; k0 < K; k0 += 4) {
        const int ka = k0 + 2 * kh;
        v2f a, b;
        a.x = aLDS[nl * aStride + ka];
        a.y = aLDS[nl * aStride + ka + 1];
        b.x = inb ? B[ka * ldB + col] : 0.0f;
        b.y = inb ? B[(ka + 1) * ldB + col] : 0.0f;
        acc = __builtin_amdgcn_wmma_f32_16x16x4_f32(false, a, false, b,
                                                    (short)0, acc, false, false);
    }
    return acc;
}

// Store C tile (ISA C layout: VGPR v -> row v + 8*(lane>=16), col = lane&15) to LDS,
// adding per-column bias and optional relu.
__device__ __forceinline__ void store_tile(float* __restrict__ outLDS, int stride, v8f c,
                                           int lane, int colBase, int maxCol,
                                           float bias, bool doRelu) {
    const int nl = lane & 15;
    const int kh = lane >> 4;
    const int col = colBase + nl;
    if (col < maxCol) {
#pragma unroll
        for (int v = 0; v < 8; ++v) {
            float val = c[v] + bias;
            if (doRelu) val = fmaxf(val, 0.0f);
            outLDS[(v + 8 * kh) * stride + col] = val;
        }
    }
}

__global__ __launch_bounds__(BLOCK) void offsets_sdf_fused(
    const float* __restrict__ points, const float* __restrict__ table,
    const float* __restrict__ W0, const float* __restrict__ b0,
    const float* __restrict__ W1, const float* __restrict__ b1,
    const float* __restrict__ W2, const float* __restrict__ b2,
    const float* __restrict__ W3, const float* __restrict__ b3,
    const float* __restrict__ eW0, const float* __restrict__ eb0,
    const float* __restrict__ eW1, const float* __restrict__ eb1,
    float* __restrict__ out) {
    __shared__ float encS[WAVES][32][ENC_STRIDE];
    __shared__ float hA[WAVES][16][H_STRIDE];
    __shared__ float hB[WAVES][16][H_STRIDE];

    const int wave = threadIdx.x >> 5;
    const int lane = threadIdx.x & 31;
    const int nl   = lane & 15;
    const int kh   = lane >> 4;
    const int waveBase = blockIdx.x * BLOCK + wave * 32;

    // ---------- hash encoding: one point per lane, staged to LDS ----------
    {
        const int p = waveBase + lane;
        const float x = points[p * 3 + 0] * 0.5f + 0.5f;  // /BB + 0.5, BB=2
        const float y = points[p * 3 + 1] * 0.5f + 0.5f;
        const float z = points[p * 3 + 2] * 0.5f + 0.5f;
        for (int l = 0; l < L_LEVELS; ++l) {
            // RES = floor(16 * exp(l*ln(128)/23)) = floor(16 * 2^(7l/23))
            const float res = floorf(16.0f * exp2f((7.0f / 23.0f) * (float)l));
            const float sx = x * res, sy = y * res, sz = z * res;
            const float fx = floorf(sx), fy = floorf(sy), fz = floorf(sz);
            const float tx = sx - fx, ty = sy - fy, tz = sz - fz;
            const unsigned cx = (unsigned)(int)fx;
            const unsigned cy = (unsigned)(int)fy;
            const unsigned cz = (unsigned)(int)fz;
            float f0 = 0.0f, f1 = 0.0f;
            const float* tl = table + (size_t)l * (size_t)T_SIZE * 2u;
#pragma unroll
            for (int c = 0; c < 8; ++c) {
                const unsigned ux = cx + (c & 1);
                const unsigned uy = cy + ((c >> 1) & 1);
                const unsigned uz = cz + ((c >> 2) & 1);
                const unsigned idx =
                    ((ux * 1u) ^ (uy * 2654435761u) ^ (uz * 805459861u)) & (T_SIZE - 1u);
                const float wx = (c & 1) ? tx : 1.0f - tx;
                const float wy = ((c >> 1) & 1) ? ty : 1.0f - ty;
                const float wz = ((c >> 2) & 1) ? tz : 1.0f - tz;
                const float w = wx * wy * wz;
                const float2 f = *(const float2*)(tl + (size_t)idx * 2u);
                f0 += w * f.x;
                f1 += w * f.y;
            }
            encS[wave][lane][2 * l + 0] = f0;
            encS[wave][lane][2 * l + 1] = f1;
        }
    }
    __syncthreads();

    float* const enc0  = &encS[wave][0][0];
    float* const bufA  = &hA[wave][0][0];
    float* const bufB  = &hB[wave][0][0];
    float* const sdfs  = out;
    float* const offs  = out + (size_t)3 * N_PTS;
    float* const geomO = out + (size_t)6 * N_PTS;

    // ---------- two 16-point WMMA tiles per wave ----------
    for (int t = 0; t < 2; ++t) {
        const float* aEnc = enc0 + (t * 16) * ENC_STRIDE;

        // layer0: 48 -> 32, relu -> bufA
        for (int nt = 0; nt < 2; ++nt) {
            v8f c = gemm_tile_f32(aEnc, ENC_STRIDE, 48, W0, 32, nt * 16, 32, lane);
            store_tile(bufA, H_STRIDE, c, lane, nt * 16, 32, b0[nt * 16 + nl], true);
        }
        __syncthreads();
        // layer1: 32 -> 32, relu -> bufB
        for (int nt = 0; nt < 2; ++nt) {
            v8f c = gemm_tile_f32(bufA, H_STRIDE, 32, W1, 32, nt * 16, 32, lane);
            store_tile(bufB, H_STRIDE, c, lane, nt * 16, 32, b1[nt * 16 + nl], true);
        }
        __syncthreads();
        // layer2: 32 -> 32, relu -> bufA
        for (int nt = 0; nt < 2; ++nt) {
            v8f c = gemm_tile_f32(bufB, H_STRIDE, 32, W2, 32, nt * 16, 32, lane);
            store_tile(bufA, H_STRIDE, c, lane, nt * 16, 32, b2[nt * 16 + nl], true);
        }
        __syncthreads();
        // layer3: 32 -> 33 (no relu), cols padded to 3 N-tiles -> bufB (pred)
        for (int nt = 0; nt < 3; ++nt) {
            const int col = nt * 16 + nl;
            const float bias = (col < 33) ? b3[col] : 0.0f;
            v8f c = gemm_tile_f32(bufA, H_STRIDE, 32, W3, 33, nt * 16, 33, lane);
            store_tile(bufB, H_STRIDE, c, lane, nt * 16, 33, bias, false);
        }
        __syncthreads();

        // eps head: geom = pred cols 1..32 (A base = bufB+1)
        float epsv[2];
        for (int k = 0; k < 2; ++k) {
            for (int nt = 0; nt < 2; ++nt) {
                v8f c = gemm_tile_f32(bufB + 1, H_STRIDE, 32, eW0 + k * 32 * 32, 32,
                                      nt * 16, 32, lane);
                store_tile(bufA, H_STRIDE, c, lane, nt * 16, 32,
                           eb0[k * 32 + nt * 16 + nl], true);
            }
            __syncthreads();
            // 32->1 projection: half-dot per lane + cross-half shuffle reduce
            float part = 0.0f;
#pragma unroll
            for (int j = 0; j < 16; ++j)
                part += bufA[nl * H_STRIDE + kh * 16 + j] * eW1[k * 32 + kh * 16 + j];
            part += __shfl_xor(part, 16, 32);
            epsv[k] = part + eb1[k];
            __syncthreads();  // WAR before bufA reuse
        }

        // epilogue (per point nl of this tile)
        const float inner_eps = softplusf(epsv[1]);
        const float outer_eps = -softplusf(epsv[0]);
        const int n = waveBase + t * 16 + nl;
        if (lane < 16) {
            const float sdf = bufB[nl * H_STRIDE + 0];
            sdfs[(size_t)n * 3 + 0] = sdf + (inner_eps + MIN_OFF);
            sdfs[(size_t)n * 3 + 1] = sdf;
            sdfs[(size_t)n * 3 + 2] = sdf + (outer_eps - MIN_OFF);
            offs[(size_t)n * 3 + 0] = inner_eps;
            offs[(size_t)n * 3 + 1] = 0.0f;
            offs[(size_t)n * 3 + 2] = outer_eps;
        }
#pragma unroll
        for (int j = 0; j < 16; ++j)
            geomO[(size_t)n * 32 + kh * 16 + j] = bufB[nl * H_STRIDE + 1 + kh * 16 + j];
        __syncthreads();  // bufB reused by next tile
    }
}

extern "C" void kernel_launch(void* const* d_in, const int* in_sizes, int n_in,
                              void* d_out, int out_size, void* d_ws, size_t ws_size,
                              hipStream_t stream) {
    (void)in_sizes; (void)n_in; (void)out_size; (void)d_ws; (void)ws_size;
    const float* points = (const float*)d_in[0];
    const float* table  = (const float*)d_in[1];
    const float* W0 = (const float*)d_in[2];
    const float* b0 = (const float*)d_in[3];
    const float* W1 = (const float*)d_in[4];
    const float* b1 = (const float*)d_in[5];
    const float* W2 = (const float*)d_in[6];
    const float* b2 = (const float*)d_in[7];
    const float* W3 = (const float*)d_in[8];
    const float* b3 = (const float*)d_in[9];
    const float* eW0 = (const float*)d_in[10];
    const float* eb0 = (const float*)d_in[11];
    const float* eW1 = (const float*)d_in[12];
    const float* eb1 = (const float*)d_in[13];
    float* out = (float*)d_out;

    dim3 grid(N_PTS / BLOCK);
    offsets_sdf_fused<<<grid, BLOCK, 0, stream>>>(points, table, W0, b0, W1, b1,
                                                  W2, b2, W3, b3, eW0, eb0, eW1, eb1, out);
}